// Mamba2Model_46712064311929
// MI455X (gfx1250) — compile-verified
//
#include <hip/hip_runtime.h>
#include <math.h>

// ---------------- model constants (match reference) ----------------
#define BATCH   2
#define SEQ     1024
#define DM      768      // d_model
#define DI      1536     // d_inner
#define DS      16       // d_state
#define NLAYERS 6
#define VOCAB   32000
#define BT      (BATCH*SEQ)   // 2048 rows

typedef float v2f __attribute__((ext_vector_type(2)));
typedef float v4f __attribute__((ext_vector_type(4)));
typedef float v8f __attribute__((ext_vector_type(8)));

__device__ __forceinline__ float clip100f(float v) {
    return fminf(fmaxf(v, -100.f), 100.f);   // -> v_med3 where possible
}
__device__ __forceinline__ float sigmoidf_dev(float v) {
    return 1.f / (1.f + expf(-v));
}

// ---------------------------------------------------------------
// fp32 WMMA GEMM:  C[M,N] = A[M,K] * W[N,K]^T + bias[N]  (+ res[M,N])
//
// V_WMMA_F32_16X16X4_F32; wave computes (MT*16) x (NT*16); 4 waves per
// block arranged WY x WX (WY = 4/WX). Block tile: (WY*MT*16) x (WX*NT*16).
// Requires M, N divisible by block tile, K % 4 == 0.
//
// Fragment layout (ISA 7.12.2, 32-bit A 16x4 / B 4x16): both A and B
// fragments are one float2 load along K per lane:
//   lane l<16  -> {row(l),  k, k+1},  lane l>=16 -> {row(l-16), k+2, k+3}
// C/D: VGPR v, lanes 0-15 -> row +v, lanes 16-31 -> row +v+8, col = l%16.
//
// Software pipelined: fragments for k+4 are loaded before the k-step
// WMMAs issue, so s_wait_loadcnt waits only on older loads and the new
// loads overlap WMMA execution. Addresses are scalar-base + 32-bit
// per-lane element offsets (saddr + voffset form, no 64-bit VALU adds).
// ---------------------------------------------------------------
template <int MT, int NT, int WX, bool ADD_RES>
__global__ __launch_bounds__(128)
void wmma_gemm_kernel(const float* __restrict__ A,
                      const float* __restrict__ W,
                      const float* __restrict__ bias,
                      const float* __restrict__ res,
                      float* __restrict__ C,
                      int M, int N, int K)
{
    constexpr int WY = 4 / WX;
    const int lane = threadIdx.x & 31;
    const int wid  = threadIdx.x >> 5;
    const int wy   = wid / WX;
    const int wx   = wid % WX;
    const int m0   = blockIdx.x * (WY * MT * 16) + wy * MT * 16;
    const int n0   = blockIdx.y * (WX * NT * 16) + wx * NT * 16;
    const int half = lane >> 4;                 // 0 or 1
    const int l16  = lane & 15;

    unsigned aoff[MT], woff[NT];
#pragma unroll
    for (int i = 0; i < MT; ++i)
        aoff[i] = (unsigned)(m0 + i * 16 + l16) * (unsigned)K + (unsigned)(half << 1);
#pragma unroll
    for (int j = 0; j < NT; ++j)
        woff[j] = (unsigned)(n0 + j * 16 + l16) * (unsigned)K + (unsigned)(half << 1);

    v8f acc[MT][NT];
#pragma unroll
    for (int i = 0; i < MT; ++i)
#pragma unroll
        for (int j = 0; j < NT; ++j)
#pragma unroll
            for (int v = 0; v < 8; ++v) acc[i][j][v] = 0.f;

    // prologue: fragments for k = 0
    v2f a_cur[MT], b_cur[NT];
#pragma unroll
    for (int i = 0; i < MT; ++i) a_cur[i] = *(const v2f*)(A + aoff[i]);
#pragma unroll
    for (int j = 0; j < NT; ++j) b_cur[j] = *(const v2f*)(W + woff[j]);

    for (int k = 4; k < K; k += 4) {
        v2f a_nxt[MT], b_nxt[NT];
#pragma unroll
        for (int i = 0; i < MT; ++i) a_nxt[i] = *(const v2f*)(A + aoff[i] + (unsigned)k);
#pragma unroll
        for (int j = 0; j < NT; ++j) b_nxt[j] = *(const v2f*)(W + woff[j] + (unsigned)k);

#pragma unroll
        for (int i = 0; i < MT; ++i)
#pragma unroll
            for (int j = 0; j < NT; ++j)
                acc[i][j] = __builtin_amdgcn_wmma_f32_16x16x4_f32(
                    false, a_cur[i], false, b_cur[j], (short)0, acc[i][j],
                    false, false);

#pragma unroll
        for (int i = 0; i < MT; ++i) a_cur[i] = a_nxt[i];
#pragma unroll
        for (int j = 0; j < NT; ++j) b_cur[j] = b_nxt[j];
    }

    // epilogue WMMAs for the last k-step
#pragma unroll
    for (int i = 0; i < MT; ++i)
#pragma unroll
        for (int j = 0; j < NT; ++j)
            acc[i][j] = __builtin_amdgcn_wmma_f32_16x16x4_f32(
                false, a_cur[i], false, b_cur[j], (short)0, acc[i][j],
                false, false);

    // store with bias (+ optional residual), C-layout per ISA
#pragma unroll
    for (int i = 0; i < MT; ++i)
#pragma unroll
        for (int j = 0; j < NT; ++j) {
            const int n    = n0 + j * 16 + l16;
            const float bv = bias ? bias[n] : 0.f;
#pragma unroll
            for (int v = 0; v < 8; ++v) {
                const int m = m0 + i * 16 + v + half * 8;
                const unsigned off = (unsigned)m * (unsigned)N + (unsigned)n;
                float val = acc[i][j][v] + bv;
                if (ADD_RES) val += res[off];
                C[off] = val;
            }
        }
}

// ---------------- embedding gather ----------------
__global__ void embed_kernel(const int* __restrict__ ids,
                             const float* __restrict__ emb,
                             float* __restrict__ x)
{
    const int idx = blockIdx.x * blockDim.x + threadIdx.x;   // over BT*DM exactly
    const int d  = idx % DM;
    const int bt = idx / DM;
    x[idx] = emb[(size_t)ids[bt] * DM + d];
}

// ---------------- rmsnorm (reference semantics) ----------------
// norm = ||x||_2 / sqrt(D);  out = w * x / (norm + 1e-8)
__global__ __launch_bounds__(256)
void rmsnorm_kernel(const float* __restrict__ x,
                    const float* __restrict__ w,
                    float* __restrict__ out, int D)
{
    __shared__ float red[8];
    __shared__ float s_scale;
    const int row = blockIdx.x;
    const float* xr = x + (size_t)row * D;

    float ss = 0.f;
    for (int i = threadIdx.x; i < D; i += 256) { float v = xr[i]; ss += v * v; }
    for (int off = 16; off > 0; off >>= 1) ss += __shfl_down(ss, off, 32);
    const int lane = threadIdx.x & 31, wid = threadIdx.x >> 5;
    if (lane == 0) red[wid] = ss;
    __syncthreads();
    if (threadIdx.x == 0) {
        float t = 0.f;
#pragma unroll
        for (int i = 0; i < 8; ++i) t += red[i];
        const float norm = sqrtf(t / (float)D);
        s_scale = 1.f / (norm + 1e-8f);
    }
    __syncthreads();
    const float sc = s_scale;
    for (int i = threadIdx.x; i < D; i += 256)
        out[(size_t)row * D + i] = w[i] * xr[i] * sc;
}

// ---------------- causal depthwise conv (K=4) + SiLU ----------------
// xz: [BT, 2*DI]; uses first DI columns. xs: [BT, DI]
__global__ void conv_silu_kernel(const float* __restrict__ xz,
                                 const float* __restrict__ cw,   // [DI,1,4]
                                 const float* __restrict__ cb,   // [DI]
                                 float* __restrict__ xs)
{
    const int idx = blockIdx.x * blockDim.x + threadIdx.x;       // over BT*DI exactly
    const int c  = idx % DI;
    const int bt = idx / DI;
    const int t  = bt % SEQ;

    float acc = cb[c];
#pragma unroll
    for (int j = 0; j < 4; ++j) {
        const int tt = t - 3 + j;
        if (tt >= 0)
            acc += cw[c * 4 + j] * xz[(size_t)(bt - (3 - j)) * (2 * DI) + c];
    }
    xs[idx] = acc * sigmoidf_dev(acc);    // silu
}

// ---------------- sequential SSM scan + output gating ----------------
// One thread per (batch, channel); 16 states in registers as 8x v2f so
// state updates lower to packed V_PK_FMA_F32 (2 FMA/instr, halves the
// serial chain). B/C rows are broadcast v4f loads. global_prefetch_b8
// issued 8 timesteps ahead hides load latency inside the serial loop.
__global__ __launch_bounds__(256)
void scan_gate_kernel(const float* __restrict__ xs,      // [BT, DI] (post conv/silu)
                      const float* __restrict__ xz,      // [BT, 2*DI] (z in cols DI..)
                      const float* __restrict__ A_log,   // [DI, DS]
                      const float* __restrict__ Dp,      // [DI]
                      const float* __restrict__ Bm,      // [BT, DS]
                      const float* __restrict__ Cm,      // [BT, DS]
                      float* __restrict__ ys)            // [BT, DI]
{
    const int c = blockIdx.x * blockDim.x + threadIdx.x;  // 0..DI-1
    const int b = blockIdx.y;

    v2f A2[8], h2[8];
#pragma unroll
    for (int q = 0; q < 8; ++q) {
#pragma unroll
        for (int e = 0; e < 2; ++e) {
            float al = A_log[(size_t)c * DS + q * 2 + e];
            al = fminf(fmaxf(al, -10.f), 2.f);
            A2[q][e] = -expf(al);
        }
        h2[q][0] = 0.f; h2[q][1] = 0.f;
    }
    const float dpc = Dp[c];
    const v2f vlo = {-100.f, -100.f};
    const v2f vhi = { 100.f,  100.f};

    for (int t = 0; t < SEQ; ++t) {
        const size_t bt = (size_t)b * SEQ + t;

        if (t + 8 < SEQ) {   // uniform branch; prefetch 8 steps ahead
            const size_t bp = bt + 8;
            __builtin_prefetch(xs + bp * DI + c, 0, 3);
            __builtin_prefetch(Bm + bp * DS, 0, 3);
            __builtin_prefetch(Cm + bp * DS, 0, 3);
            __builtin_prefetch(xz + bp * (size_t)(2 * DI) + DI + c, 0, 3);
        }

        const float xv = xs[bt * DI + c];
        const v4f* Bv = (const v4f*)(Bm + bt * DS);
        const v4f* Cv = (const v4f*)(Cm + bt * DS);
        const v2f xv2 = {xv, xv};

        v2f yv = {0.f, 0.f};
#pragma unroll
        for (int p = 0; p < 4; ++p) {
            const v4f Bq = Bv[p];
            const v4f Cq = Cv[p];
            const v2f B0 = {Bq[0], Bq[1]}, B1 = {Bq[2], Bq[3]};
            const v2f C0 = {Cq[0], Cq[1]}, C1 = {Cq[2], Cq[3]};

            v2f hn0 = __builtin_elementwise_fma(h2[2*p+0], A2[2*p+0], xv2 * B0);
            v2f hn1 = __builtin_elementwise_fma(h2[2*p+1], A2[2*p+1], xv2 * B1);
            hn0 = __builtin_elementwise_min(__builtin_elementwise_max(hn0, vlo), vhi);
            hn1 = __builtin_elementwise_min(__builtin_elementwise_max(hn1, vlo), vhi);
            h2[2*p+0] = hn0;
            h2[2*p+1] = hn1;
            yv = __builtin_elementwise_fma(hn0, C0, yv);
            yv = __builtin_elementwise_fma(hn1, C1, yv);
        }
        float y = yv[0] + yv[1] + dpc * xv;
        y = clip100f(y);
        const float z = xz[bt * (size_t)(2 * DI) + DI + c];
        ys[bt * DI + c] = y * sigmoidf_dev(z);
    }
}

// ---------------------------------------------------------------
extern "C" void kernel_launch(void* const* d_in, const int* in_sizes, int n_in,
                              void* d_out, int out_size, void* d_ws, size_t ws_size,
                              hipStream_t stream)
{
    const int*   ids     = (const int*)  d_in[0];
    const float* emb     = (const float*)d_in[1];
    const float* norm_w  = (const float*)d_in[2];
    const float* in_w    = (const float*)d_in[3];
    const float* in_b    = (const float*)d_in[4];
    const float* conv_w  = (const float*)d_in[5];
    const float* conv_b  = (const float*)d_in[6];
    const float* A_log   = (const float*)d_in[7];
    const float* Dp      = (const float*)d_in[8];
    const float* Bp_w    = (const float*)d_in[9];
    const float* Bp_b    = (const float*)d_in[10];
    const float* Cp_w    = (const float*)d_in[11];
    const float* Cp_b    = (const float*)d_in[12];
    const float* out_w   = (const float*)d_in[13];
    const float* out_b   = (const float*)d_in[14];
    const float* fnorm_w = (const float*)d_in[15];
    const float* lm_w    = (const float*)d_in[16];
    const float* lm_b    = (const float*)d_in[17];
    float* logits = (float*)d_out;

    // workspace layout (floats), ~64 MB total
    float* ws = (float*)d_ws;
    float* x   = ws;                               // BT*DM residual stream
    float* xn  = x  + (size_t)BT * DM;             // BT*DM normalized
    float* xz  = xn + (size_t)BT * DM;             // BT*2*DI in_proj output
    float* xs  = xz + (size_t)BT * 2 * DI;         // BT*DI conv+silu
    float* Bm  = xs + (size_t)BT * DI;             // BT*DS
    float* Cm  = Bm + (size_t)BT * DS;             // BT*DS
    float* ys  = Cm + (size_t)BT * DS;             // BT*DI

    // x = emb[input_ids]
    embed_kernel<<<(BT * DM) / 256, 256, 0, stream>>>(ids, emb, x);

    for (int l = 0; l < NLAYERS; ++l) {
        // xn = rmsnorm(x, norm_w[l])
        rmsnorm_kernel<<<BT, 256, 0, stream>>>(x, norm_w + (size_t)l * DM, xn, DM);

        // xz = xn @ in_w[l]^T + in_b[l]    [BT, 2*DI]; block tile 64x128
        {
            dim3 g(BT / 64, (2 * DI) / 128);
            wmma_gemm_kernel<2, 4, 2, false><<<g, 128, 0, stream>>>(
                xn, in_w + (size_t)l * 2 * DI * DM, in_b + (size_t)l * 2 * DI,
                nullptr, xz, BT, 2 * DI, DM);
        }

        // xs = silu(causal_dwconv(xz[:, :DI]))
        conv_silu_kernel<<<(BT * DI) / 256, 256, 0, stream>>>(
            xz, conv_w + (size_t)l * DI * 4, conv_b + (size_t)l * DI, xs);

        // Bm = xs @ Bp_w[l]^T + Bp_b[l];  Cm likewise   [BT, 16]; tile 128x16
        {
            dim3 g(BT / 128, 1);
            wmma_gemm_kernel<2, 1, 1, false><<<g, 128, 0, stream>>>(
                xs, Bp_w + (size_t)l * DS * DI, Bp_b + (size_t)l * DS,
                nullptr, Bm, BT, DS, DI);
            wmma_gemm_kernel<2, 1, 1, false><<<g, 128, 0, stream>>>(
                xs, Cp_w + (size_t)l * DS * DI, Cp_b + (size_t)l * DS,
                nullptr, Cm, BT, DS, DI);
        }

        // ys = gated SSM scan
        scan_gate_kernel<<<dim3(DI / 256, BATCH), 256, 0, stream>>>(
            xs, xz, A_log + (size_t)l * DI * DS, Dp + (size_t)l * DI, Bm, Cm, ys);

        // x = ys @ out_w[l]^T + out_b[l] + x   (fused residual, in place)
        {
            dim3 g(BT / 64, DM / 128);
            wmma_gemm_kernel<2, 4, 2, true><<<g, 128, 0, stream>>>(
                ys, out_w + (size_t)l * DM * DI, out_b + (size_t)l * DM,
                x, x, BT, DM, DI);
        }
    }

    // final norm + LM head
    rmsnorm_kernel<<<BT, 256, 0, stream>>>(x, fnorm_w, xn, DM);
    {
        dim3 g(BT / 64, VOCAB / 128);
        wmma_gemm_kernel<2, 4, 2, false><<<g, 128, 0, stream>>>(
            xn, lm_w, lm_b, nullptr, logits, BT, VOCAB, DM);
    }
}